// SpatialFrequencyLoss_49108656063011
// MI455X (gfx1250) — compile-verified
//
#include <hip/hip_runtime.h>
#include <hip/hip_bf16.h>

// ---------------------------------------------------------------------------
// SpatialFrequencyLoss on gfx1250 (MI455X), wave32.
//
//   d[b,h,w]  = sum_c [(p[c,h,w-1]-p)^2 + (p[c,h-1,w]-p)^2]   (zero pad)
//             - sum_c [(t[c,h,w-1]-t)^2 + (t[c,h-1,w]-t)^2]
//   s1 = horizontal 11-wide box sum of d     (WMMA: D * bandW)
//   s2 = vertical   11-tall box sum of s1    (WMMA: bandM * S1)
//   out = mean |s2|
//
// Box conv is linear -> difference taken BEFORE the filter; only one field.
// ---------------------------------------------------------------------------

typedef __attribute__((ext_vector_type(2))) float v2f;
typedef __attribute__((ext_vector_type(8))) float v8f;

#define NIMG 16
#define IMG_H 512
#define IMG_W 512
#define NCH 3
#define RAD 5
// tiles: 16 x (512/16) x (512/16) = 16384 tiles; 8 waves/block -> 2048 blocks
#define TILES_PER_IMG ((IMG_H / 16) * (IMG_W / 16))
#define N_TILES (NIMG * TILES_PER_IMG)
#define WAVES_PER_BLOCK 8
#define N_TILE_BLOCKS (N_TILES / WAVES_PER_BLOCK)

// ---------------------------------------------------------------------------
// Kernel 1: squared-gradient difference field (the only HBM-heavy pass).
// ---------------------------------------------------------------------------
__global__ void sfl_grad_diff(const float* __restrict__ pred,
                              const float* __restrict__ targ,
                              float* __restrict__ d) {
    int idx = blockIdx.x * blockDim.x + threadIdx.x;   // exact grid, no guard
    int w = idx & (IMG_W - 1);
    int h = (idx >> 9) & (IMG_H - 1);
    int b = idx >> 18;

    float acc = 0.f;
#pragma unroll
    for (int c = 0; c < NCH; ++c) {
        size_t base = (((size_t)(b * NCH + c)) * IMG_H + h) * IMG_W + w;
        float p  = pred[base];
        float pl = (w > 0) ? pred[base - 1]     : 0.f;
        float pu = (h > 0) ? pred[base - IMG_W] : 0.f;
        float t  = targ[base];
        float tl = (w > 0) ? targ[base - 1]     : 0.f;
        float tu = (h > 0) ? targ[base - IMG_W] : 0.f;
        float dph = pl - p, dpv = pu - p;
        float dth = tl - t, dtv = tu - t;
        acc += dph * dph + dpv * dpv - dth * dth - dtv * dtv;
    }
    d[(size_t)b * (IMG_H * IMG_W) + h * IMG_W + w] = acc;
}

// ---------------------------------------------------------------------------
// Kernel 2: horizontal 11-wide box sum via V_WMMA_F32_16X16X4_F32.
// One wave = one 16x16 tile. T(16x16) = D(16x28) x W(28x16), 7 K=4 steps.
// A layout (16x4 f32): lane&15 = M; VGPR0 = {K0 | K2}, VGPR1 = {K1 | K3}.
// B layout (4x16 f32): lane&15 = N; VGPR0 = {K0 | K2}, VGPR1 = {K1 | K3}.
// C/D layout: VGPR v -> M = v + 8*(lane>>4), N = lane&15.
// ---------------------------------------------------------------------------
__global__ void sfl_box_h(const float* __restrict__ d, float* __restrict__ s1) {
    int wave = threadIdx.x >> 5;
    int lane = threadIdx.x & 31;
    int n    = lane & 15;          // M for A-side, N for B/C-side
    int half = lane >> 4;

    int tile = blockIdx.x * WAVES_PER_BLOCK + wave;
    int b  = tile >> 10;                       // / (32*32)
    int tr = (tile >> 5) & 31;
    int tc = tile & 31;
    int r0 = tr * 16;
    int c0 = tc * 16;

    const float* dimg = d + (size_t)b * (IMG_H * IMG_W);

    v8f acc = {};
#pragma unroll
    for (int kc = 0; kc < 28; kc += 4) {
        int j0 = kc + 2 * half;                // K indices j0, j0+1
        int col0 = c0 - RAD + j0;
        int col1 = col0 + 1;
        int row  = r0 + n;

        v2f a, bm;
        a.x = (col0 >= 0 && col0 < IMG_W) ? dimg[row * IMG_W + col0] : 0.f;
        a.y = (col1 >= 0 && col1 < IMG_W) ? dimg[row * IMG_W + col1] : 0.f;
        // band: output col n sums input j in [n, n+10]; zero for padded K
        bm.x = (j0     >= n && j0     <= n + 2 * RAD) ? 1.f : 0.f;
        bm.y = (j0 + 1 >= n && j0 + 1 <= n + 2 * RAD) ? 1.f : 0.f;

        acc = __builtin_amdgcn_wmma_f32_16x16x4_f32(
            false, a, false, bm, (short)0, acc, false, false);
    }

    float* s1img = s1 + (size_t)b * (IMG_H * IMG_W);
#pragma unroll
    for (int v = 0; v < 8; ++v)
        s1img[(r0 + v + 8 * half) * IMG_W + (c0 + n)] = acc[v];
}

// ---------------------------------------------------------------------------
// Kernel 3: vertical 11-tall box sum via WMMA + |.| + deterministic
// per-block reduction into partials[blockIdx.x].
// T(16x16) = bandM(16x28) x S1rows(28x16).
// ---------------------------------------------------------------------------
__global__ void sfl_box_v_reduce(const float* __restrict__ s1,
                                 float* __restrict__ partials) {
    int wave = threadIdx.x >> 5;
    int lane = threadIdx.x & 31;
    int n    = lane & 15;
    int half = lane >> 4;

    int tile = blockIdx.x * WAVES_PER_BLOCK + wave;
    int b  = tile >> 10;
    int tr = (tile >> 5) & 31;
    int tc = tile & 31;
    int r0 = tr * 16;
    int c0 = tc * 16;

    const float* s1img = s1 + (size_t)b * (IMG_H * IMG_W);

    v8f acc = {};
#pragma unroll
    for (int kc = 0; kc < 28; kc += 4) {
        int j0 = kc + 2 * half;
        int rin0 = r0 - RAD + j0;
        int rin1 = rin0 + 1;

        v2f a, bm;
        // band: output row m=n sums input j in [m, m+10]; zero for padded K
        a.x = (j0     >= n && j0     <= n + 2 * RAD) ? 1.f : 0.f;
        a.y = (j0 + 1 >= n && j0 + 1 <= n + 2 * RAD) ? 1.f : 0.f;
        bm.x = (rin0 >= 0 && rin0 < IMG_H) ? s1img[rin0 * IMG_W + (c0 + n)] : 0.f;
        bm.y = (rin1 >= 0 && rin1 < IMG_H) ? s1img[rin1 * IMG_W + (c0 + n)] : 0.f;

        acc = __builtin_amdgcn_wmma_f32_16x16x4_f32(
            false, a, false, bm, (short)0, acc, false, false);
    }

    float s = 0.f;
#pragma unroll
    for (int v = 0; v < 8; ++v) s += fabsf(acc[v]);

    __shared__ float red[256];
    red[threadIdx.x] = s;
    __syncthreads();
    if (threadIdx.x == 0) {
        float tot = 0.f;
        for (int i = 0; i < 256; ++i) tot += red[i];   // fixed order: deterministic
        partials[blockIdx.x] = tot;
    }
}

// ---------------------------------------------------------------------------
// Kernel 4: single-block deterministic final reduction + mean.
// ---------------------------------------------------------------------------
__global__ void sfl_final(const float* __restrict__ partials, float* __restrict__ out) {
    __shared__ float red[256];
    float s = 0.f;
    for (int i = threadIdx.x; i < N_TILE_BLOCKS; i += 256) s += partials[i];
    red[threadIdx.x] = s;
    __syncthreads();
    if (threadIdx.x == 0) {
        float tot = 0.f;
        for (int i = 0; i < 256; ++i) tot += red[i];
        out[0] = tot / (float)((size_t)NIMG * IMG_H * IMG_W);
    }
}

// ---------------------------------------------------------------------------
extern "C" void kernel_launch(void* const* d_in, const int* in_sizes, int n_in,
                              void* d_out, int out_size, void* d_ws, size_t ws_size,
                              hipStream_t stream) {
    const float* pred = (const float*)d_in[0];
    const float* targ = (const float*)d_in[1];
    float* out = (float*)d_out;

    const size_t npix = (size_t)NIMG * IMG_H * IMG_W;   // 4,194,304
    float* dbuf   = (float*)d_ws;                       // 16.78 MB
    float* s1buf  = dbuf + npix;                        // 16.78 MB
    float* parts  = s1buf + npix;                       // 8 KB

    sfl_grad_diff<<<(int)(npix / 256), 256, 0, stream>>>(pred, targ, dbuf);
    sfl_box_h<<<N_TILE_BLOCKS, 256, 0, stream>>>(dbuf, s1buf);
    sfl_box_v_reduce<<<N_TILE_BLOCKS, 256, 0, stream>>>(s1buf, parts);
    sfl_final<<<1, 256, 0, stream>>>(parts, out);
}